// RegularSHT_335007449422
// MI455X (gfx1250) — compile-verified
//
#include <hip/hip_runtime.h>

// ---------------------------------------------------------------------------
// Regular SHT on gfx1250 via V_WMMA_F32_16X16X4_F32 with LDS-staged B tiles.
//   stage 0: tw[mcs=256][n=512]; mcs<128 -> cos(2*pi*m*n/512), else -sin.
//   stage 1: per (bte,k):  F[plane][m][k][c] = quad_w[k] * (tw x data_slice)
//            GEMM  [256 x 64] = [256 x 512] * [512 x 64]   (K staged in LDS)
//   stage 2: per (m,plane,chalf): out = legendre_m x F_m
//            GEMM  [128 x 128] = [128 x 256] * [256 x 128] (K staged in LDS)
// Output layout [b,t,e,l,m,v] complex64 (re/im interleaved floats).
// ---------------------------------------------------------------------------

#define NLAT 256
#define NLON 512
#define LMAX 128
#define MMAX 128
#define NBTE 4      // B*T*E = 2*2*1
#define NV   64
#define NC   256    // NBTE*NV channels
#define KT   32     // K-tile staged through LDS
#define LPAD 2      // LDS row pad (row stride 34 words -> conflict-free frags)

typedef __attribute__((ext_vector_type(2))) float v2f;
typedef __attribute__((ext_vector_type(4))) float v4f;
typedef __attribute__((ext_vector_type(8))) float v8f;

__device__ __forceinline__ v8f wmma_f32(v2f a, v2f b, v8f c) {
  // D = A(16x4,f32) * B(4x16,f32) + C(16x16,f32)
  return __builtin_amdgcn_wmma_f32_16x16x4_f32(
      /*neg_a=*/false, a, /*neg_b=*/false, b,
      /*c_mod=*/(short)0, c, /*reuse_a=*/false, /*reuse_b=*/false);
}

// ---------------------------------------------------------------------------
// Stage 0: twiddle table.  tw[mcs][n],  256*512 floats.
// ---------------------------------------------------------------------------
__global__ __launch_bounds__(256) void sht_twiddle_kernel(float* __restrict__ tw) {
  const int idx = blockIdx.x * 256 + threadIdx.x;    // 131072 total
  const int mcs = idx >> 9;
  const int n   = idx & (NLON - 1);
  const int m   = mcs & (MMAX - 1);
  const int r   = (m * n) & (NLON - 1);              // phase mod 2*pi
  const float ang = (float)r * (float)(6.283185307179586 / (double)NLON);
  float s, c;
  __sincosf(ang, &s, &c);
  tw[idx] = (mcs < MMAX) ? c : -s;
}

// ---------------------------------------------------------------------------
// Stage 1: longitude DFT as WMMA GEMM, data tile staged transposed in LDS.
// grid.x = NBTE*NLAT = 1024 blocks, 256 threads (8 waves).
// Block (bte,k): out[mcs=256, v=64] over K=n=512.
// Wave w owns mcs rows [32w, 32w+32): 2 M-tiles x 4 N-tiles (64 acc VGPRs).
// F layout: F[plane][m][k][c],  c = bte*64 + v  (lane-contiguous stores).
// ---------------------------------------------------------------------------
__global__ __launch_bounds__(256) void sht_dft_kernel(
    const float* __restrict__ data, const float* __restrict__ quad_w,
    const float* __restrict__ tw, float* __restrict__ F) {
  const int k    = blockIdx.x & (NLAT - 1);
  const int bte  = blockIdx.x >> 8;
  const int tid  = threadIdx.x;
  const int lane = tid & 31;
  const int wave = tid >> 5;
  const int half = lane >> 4;     // 0: K pair {0,1}, 1: K pair {2,3}
  const int lh   = lane & 15;

  const float* dbase = data + (size_t)(bte * NLAT + k) * NLON * NV;  // [n][v]
  const int mcs0 = wave * 32;

  __shared__ float sA[NV][KT + LPAD];   // data tile, transposed: [v][n]

  v8f acc[2][4] = {};

  for (int t0 = 0; t0 < NLON; t0 += KT) {
    // ---- cooperative load of 32(n) x 64(v) tile, transpose into LDS ----
    v4f r[2];
#pragma unroll
    for (int c = 0; c < 2; ++c) {
      const int f  = c * 1024 + tid * 4;     // 2048 floats total
      const int n  = f >> 6;                 // 0..31
      const int v0 = f & 63;                 // multiple of 4
      r[c] = *(const v4f*)(dbase + (size_t)(t0 + n) * NV + v0);
    }
    __syncthreads();                         // previous tile fully consumed
#pragma unroll
    for (int c = 0; c < 2; ++c) {
      const int f  = c * 1024 + tid * 4;
      const int n  = f >> 6;
      const int v0 = f & 63;
      sA[v0 + 0][n] = r[c].x;
      sA[v0 + 1][n] = r[c].y;
      sA[v0 + 2][n] = r[c].z;
      sA[v0 + 3][n] = r[c].w;
    }
    __syncthreads();                         // tile visible to all waves

    // ---- 8 WMMA K-steps over the staged tile ----
#pragma unroll
    for (int nn = 0; nn < KT; nn += 4) {
      const int n0 = t0 + nn;
      v2f a[2];
#pragma unroll
      for (int mr = 0; mr < 2; ++mr) {
        const int row = mcs0 + mr * 16 + lh;
        a[mr] = *(const v2f*)(tw + (size_t)row * NLON + n0 + half * 2);
      }
      v2f b[4];
#pragma unroll
      for (int nc = 0; nc < 4; ++nc)
        b[nc] = *(const v2f*)(&sA[nc * 16 + lh][nn + half * 2]);
#pragma unroll
      for (int mr = 0; mr < 2; ++mr)
#pragma unroll
        for (int nc = 0; nc < 4; ++nc)
          acc[mr][nc] = wmma_f32(a[mr], b[nc], acc[mr][nc]);
    }
  }

  // Fold quadrature weight (scalar per block) and store.
  const float qk = quad_w[k];
#pragma unroll
  for (int mr = 0; mr < 2; ++mr) {
#pragma unroll
    for (int nc = 0; nc < 4; ++nc) {
#pragma unroll
      for (int j = 0; j < 8; ++j) {
        const int mcs   = mcs0 + mr * 16 + j + 8 * half;  // C layout: M=j+8*half
        const int v     = nc * 16 + lh;                   // N = lane
        const int plane = mcs >> 7;
        const int m     = mcs & (MMAX - 1);
        F[((size_t)(plane * MMAX + m) * NLAT + k) * NC + bte * NV + v] =
            acc[mr][nc][j] * qk;
      }
    }
  }
}

// ---------------------------------------------------------------------------
// Stage 2: Legendre projection as WMMA GEMM, F tile staged transposed in LDS.
// grid.x = 512: m (7b) | plane (1b) | chalf (1b). 256 threads (8 waves).
// Block: out[l=128, c=128-half] over K=k=256.  Wave w owns l rows [16w,16w+16).
// ---------------------------------------------------------------------------
__global__ __launch_bounds__(256) void sht_leg_kernel(
    const float* __restrict__ legendre, const float* __restrict__ F,
    float* __restrict__ out) {
  const int m     = blockIdx.x & (MMAX - 1);
  const int plane = (blockIdx.x >> 7) & 1;
  const int chalf = blockIdx.x >> 8;
  const int tid   = threadIdx.x;
  const int lane  = tid & 31;
  const int wave  = tid >> 5;
  const int half  = lane >> 4;
  const int lh    = lane & 15;

  const int l0 = wave * 16;
  // legendre[l][m][k]: element (l,k) at l*MMAX*NLAT + m*NLAT + k
  const float* legm = legendre + (size_t)m * NLAT;
  const float* Fm   = F + (size_t)(plane * MMAX + m) * NLAT * NC;  // [k][c]

  __shared__ float sB[128][KT + LPAD];  // F tile, transposed: [c][k]

  v8f acc[8] = {};

  for (int t0 = 0; t0 < NLAT; t0 += KT) {
    // ---- cooperative load of 32(k) x 128(c) tile, transpose into LDS ----
    v4f r[4];
#pragma unroll
    for (int c = 0; c < 4; ++c) {
      const int f  = c * 1024 + tid * 4;    // 4096 floats total
      const int kk = f >> 7;                // 0..31
      const int cc = f & 127;               // multiple of 4
      r[c] = *(const v4f*)(Fm + (size_t)(t0 + kk) * NC + chalf * 128 + cc);
    }
    __syncthreads();
#pragma unroll
    for (int c = 0; c < 4; ++c) {
      const int f  = c * 1024 + tid * 4;
      const int kk = f >> 7;
      const int cc = f & 127;
      sB[cc + 0][kk] = r[c].x;
      sB[cc + 1][kk] = r[c].y;
      sB[cc + 2][kk] = r[c].z;
      sB[cc + 3][kk] = r[c].w;
    }
    __syncthreads();

    // ---- 8 WMMA K-steps over the staged tile ----
#pragma unroll
    for (int kk = 0; kk < KT; kk += 4) {
      const v2f a = *(const v2f*)(legm + (size_t)(l0 + lh) * (MMAX * NLAT) +
                                  t0 + kk + half * 2);
#pragma unroll
      for (int nc = 0; nc < 8; ++nc) {
        const v2f b = *(const v2f*)(&sB[nc * 16 + lh][kk + half * 2]);
        acc[nc] = wmma_f32(a, b, acc[nc]);
      }
    }
  }

  // out[b,t,e,l,m,v] complex interleaved:
  // idx = ((bte*LMAX + l)*MMAX + m)*NV*2 + v*2 + plane
#pragma unroll
  for (int nc = 0; nc < 8; ++nc) {
#pragma unroll
    for (int j = 0; j < 8; ++j) {
      const int l   = l0 + j + 8 * half;
      const int c   = chalf * 128 + nc * 16 + lh;
      const int bte = c >> 6;
      const int v   = c & (NV - 1);
      out[(size_t)bte * (LMAX * MMAX * NV * 2) + (size_t)l * (MMAX * NV * 2) +
          (size_t)m * (NV * 2) + v * 2 + plane] = acc[nc][j];
    }
  }
}

// ---------------------------------------------------------------------------
// Host launcher. Inputs: [0]=data f32 [2,2,1,131072,64], [1]=legendre f32
// [128,128,256], [2]=quad_w f32 [256]. Output: complex64 viewed as f32 pairs.
// Workspace: 512KB twiddle + 64MB F  (~64.5 MB total).
// ---------------------------------------------------------------------------
extern "C" void kernel_launch(void* const* d_in, const int* in_sizes, int n_in,
                              void* d_out, int out_size, void* d_ws, size_t ws_size,
                              hipStream_t stream) {
  const float* data     = (const float*)d_in[0];
  const float* legendre = (const float*)d_in[1];
  const float* quad_w   = (const float*)d_in[2];
  float* out = (float*)d_out;

  float* tw = (float*)d_ws;                 // 256*512 floats
  float* F  = tw + 256 * NLON;              // 2*128*256*256 floats

  sht_twiddle_kernel<<<dim3(512), dim3(256), 0, stream>>>(tw);
  sht_dft_kernel<<<dim3(NBTE * NLAT), dim3(256), 0, stream>>>(data, quad_w, tw, F);
  sht_leg_kernel<<<dim3(512), dim3(256), 0, stream>>>(legendre, F, out);
}